// PhysDecoder_74612171866499
// MI455X (gfx1250) — compile-verified
//
#include <hip/hip_runtime.h>
#include <math.h>

typedef __attribute__((ext_vector_type(2))) float v2f;
typedef __attribute__((ext_vector_type(8))) float v8f;

#define XSZ   256
#define NPIX  65536         // 256*256
#define NBATCH 16
#define NPTS  400000
#define LATD  10

// ws layout in floats: H[256] @ 0, a[16] @ 256, b[16] @ 272, raw images @ 512
#define WS_H   0
#define WS_A   256
#define WS_B   272
#define WS_IMG 512

// ---------------------------------------------------------------- init
__global__ void pd_init_ws(float* __restrict__ ws, int n) {
    int i = blockIdx.x * blockDim.x + threadIdx.x;
    if (i < n) ws[i] = 0.0f;
}

// ---------------------------------------------------------------- MLP GEMM via fp32 WMMA
// H(16x16) = X(16x65536) @ W1pad(65536x16); W1 cols >= LATD are zero.
// 512 waves total, each owns a K-slice of 128 (32 wmma 16x16x4 steps),
// merged into ws H with global fp32 atomics.
__global__ void pd_mlp_gemm_wmma(const float* __restrict__ x,
                                 const float* __restrict__ w1,
                                 float* __restrict__ H) {
    const int lane  = threadIdx.x & 31;
    const int lrow  = lane & 15;      // M row (A) / N col (B)
    const int half  = lane >> 4;      // K half: 0 -> K+0,1 ; 1 -> K+2,3
    const int wave  = threadIdx.x >> 5;
    const int gw    = blockIdx.x * (blockDim.x >> 5) + wave;  // 0..511
    const int kbase = gw * 128;

    const float* __restrict__ xrow = x + (size_t)lrow * NPIX;
    const int ncol = (lrow < LATD) ? lrow : 0;  // clamp to stay in-bounds
    const float nmask = (lrow < LATD) ? 1.0f : 0.0f;

    v8f c = {};
    for (int k = kbase; k < kbase + 128; k += 4) {
        const int kk = k + half * 2;
        v2f a, b;
        a.x = xrow[kk];
        a.y = xrow[kk + 1];
        b.x = nmask * w1[(size_t)kk * LATD + ncol];
        b.y = nmask * w1[(size_t)(kk + 1) * LATD + ncol];
        // 8 args: (neg_a, A, neg_b, B, c_mod, C, reuse_a, reuse_b)
        c = __builtin_amdgcn_wmma_f32_16x16x4_f32(
                false, a, false, b, (short)0, c, false, false);
    }

#pragma unroll
    for (int i = 0; i < 8; ++i) {
        const int m = i + half * 8;           // C/D layout: VGPR i -> M=i / i+8
        atomicAdd(&H[m * 16 + lrow], c[i]);
    }
}

// ---------------------------------------------------------------- MLP epilogue: relu + (10x2)
__global__ void pd_mlp_finalize(const float* __restrict__ H,
                                const float* __restrict__ b1,
                                const float* __restrict__ w2,
                                const float* __restrict__ b2,
                                float* __restrict__ ws) {
    const int b = threadIdx.x;
    if (b >= NBATCH) return;
    float acc0 = b2[0], acc1 = b2[1];
#pragma unroll
    for (int l = 0; l < LATD; ++l) {
        float h = H[b * 16 + l] + b1[l];
        h = h > 0.0f ? h : 0.0f;
        acc0 += h * w2[l * 2 + 0];
        acc1 += h * w2[l * 2 + 1];
    }
    ws[WS_A + b] = acc0;
    ws[WS_B + b] = acc1;
}

// ---------------------------------------------------------------- scatter (+ fused blur when GPB==1)
// One full 256KB fp32 image lives in LDS per workgroup (fits CDNA5 320KB/WGP).
// GPB = workgroups per batch. MERGE path: LDS image merged into ws raw images
// via global fp32 atomics, blur done by pd_blur_affine. Fallback (GPB==1):
// blur+affine fused directly from LDS.
template <int GPB, bool MERGE>
__global__ void pd_scatter(const float* __restrict__ flow,
                           const float* __restrict__ values,
                           const float* __restrict__ rotations,
                           const float* __restrict__ shifts,
                           const int*   __restrict__ inds,
                           float* __restrict__ ws,
                           float* __restrict__ out) {
    extern __shared__ float img[];   // NPIX floats = 256 KB
    const int batch = blockIdx.x / GPB;
    const int part  = blockIdx.x % GPB;
    const int t  = threadIdx.x;
    const int nt = blockDim.x;

    for (int p = t; p < NPIX; p += nt) img[p] = 0.0f;

    const float r00 = rotations[batch * 9 + 0];
    const float r01 = rotations[batch * 9 + 1];
    const float r02 = rotations[batch * 9 + 2];
    const float r10 = rotations[batch * 9 + 3];
    const float r11 = rotations[batch * 9 + 4];
    const float r12 = rotations[batch * 9 + 5];
    const float s0  = shifts[batch * 2 + 0];
    const float s1  = shifts[batch * 2 + 1];
    const float factor = 0.5f * XSZ;
    const float* __restrict__ fl = flow + (size_t)batch * NPTS * 3;

    __syncthreads();

    for (int n = part * nt + t; n < NPTS; n += GPB * nt) {
        const int i0 = inds[n * 3 + 0];
        const int i1 = inds[n * 3 + 1];
        const int i2 = inds[n * 3 + 2];
        // coords = (inds2, inds1, inds0) - 128 + flow
        const float v0 = (float)i2 - factor + fl[n * 3 + 0];
        const float v1 = (float)i1 - factor + fl[n * 3 + 1];
        const float v2 = (float)i0 - factor + fl[n * 3 + 2];
        // rotated rows 0,1; minus shift, plus factor
        const float p0 = v0 * r00 + v1 * r01 + v2 * r02 - s0 + factor;
        const float p1 = v0 * r10 + v1 * r11 + v2 * r12 - s1 + factor;
        const float amp = values[n];

        // sampling = (p1, p0): row = floor(p1), col = floor(p0)
        const float fr = floorf(p1), fc = floorf(p0);
        const int row = (int)fr, col = (int)fc;
        const float f0 = p1 - fr;      // row frac
        const float f1 = p0 - fc;      // col frac

        const float w00 = amp * (1.0f - f0) * (1.0f - f1);  // (row,   col)
        const float w10 = amp * f0 * (1.0f - f1);           // (row+1, col)
        const float w11 = amp * f0 * f1;                    // (row+1, col+1)
        const float w01 = amp * (1.0f - f0) * f1;           // (row,   col+1)

        const bool r0ok = (row >= 0) && (row < XSZ);
        const bool r1ok = (row + 1 >= 0) && (row + 1 < XSZ);
        const bool c0ok = (col >= 0) && (col < XSZ);
        const bool c1ok = (col + 1 >= 0) && (col + 1 < XSZ);
        if (r0ok && c0ok) atomicAdd(&img[row * XSZ + col], w00);
        if (r1ok && c0ok) atomicAdd(&img[(row + 1) * XSZ + col], w10);
        if (r1ok && c1ok) atomicAdd(&img[(row + 1) * XSZ + col + 1], w11);
        if (r0ok && c1ok) atomicAdd(&img[row * XSZ + col + 1], w01);
    }

    __syncthreads();

    if (MERGE) {
        float* __restrict__ raw = ws + WS_IMG + (size_t)batch * NPIX;
        for (int p = t; p < NPIX; p += nt) {
            const float v = img[p];
            if (v != 0.0f) atomicAdd(&raw[p], v);
        }
    } else {
        const float g  = expf(-0.5f);
        const float nrm = 1.0f / (1.0f + 2.0f * g);
        const float k0 = g * nrm, k1 = nrm;
        const float aa = ws[WS_A + batch];
        const float bb = ws[WS_B + batch];
        for (int p = t; p < NPIX; p += nt) {
            const int row = p >> 8, col = p & 255;
            float sum = 0.0f;
#pragma unroll
            for (int dy = -1; dy <= 1; ++dy) {
                const int rr = row + dy;
                if (rr < 0 || rr >= XSZ) continue;
                const float ky = (dy == 0) ? k1 : k0;
#pragma unroll
                for (int dx = -1; dx <= 1; ++dx) {
                    const int cc = col + dx;
                    if (cc < 0 || cc >= XSZ) continue;
                    const float kx = (dx == 0) ? k1 : k0;
                    sum += ky * kx * img[rr * XSZ + cc];
                }
            }
            out[(size_t)batch * NPIX + p] = aa * sum + bb;
        }
    }
}

// ---------------------------------------------------------------- blur + affine (merge path)
__global__ void pd_blur_affine(const float* __restrict__ ws, float* __restrict__ out) {
    const int idx = blockIdx.x * blockDim.x + threadIdx.x;
    if (idx >= NBATCH * NPIX) return;
    const int batch = idx >> 16;
    const int p = idx & (NPIX - 1);
    const int row = p >> 8, col = p & 255;

    const float g  = expf(-0.5f);
    const float nrm = 1.0f / (1.0f + 2.0f * g);
    const float k0 = g * nrm, k1 = nrm;
    const float* __restrict__ raw = ws + WS_IMG + (size_t)batch * NPIX;

    float sum = 0.0f;
#pragma unroll
    for (int dy = -1; dy <= 1; ++dy) {
        const int rr = row + dy;
        if (rr < 0 || rr >= XSZ) continue;
        const float ky = (dy == 0) ? k1 : k0;
#pragma unroll
        for (int dx = -1; dx <= 1; ++dx) {
            const int cc = col + dx;
            if (cc < 0 || cc >= XSZ) continue;
            const float kx = (dx == 0) ? k1 : k0;
            sum += ky * kx * raw[rr * XSZ + cc];
        }
    }
    out[idx] = ws[WS_A + batch] * sum + ws[WS_B + batch];
}

// ---------------------------------------------------------------- launch
extern "C" void kernel_launch(void* const* d_in, const int* in_sizes, int n_in,
                              void* d_out, int out_size, void* d_ws, size_t ws_size,
                              hipStream_t stream) {
    (void)in_sizes; (void)n_in; (void)out_size;
    const float* flow      = (const float*)d_in[0];
    const float* x         = (const float*)d_in[1];
    const float* values    = (const float*)d_in[2];
    const float* rotations = (const float*)d_in[3];
    const float* shifts    = (const float*)d_in[4];
    const float* w1        = (const float*)d_in[5];
    const float* b1        = (const float*)d_in[6];
    const float* w2        = (const float*)d_in[7];
    const float* b2        = (const float*)d_in[8];
    const int*   inds      = (const int*)d_in[9];
    float* ws  = (float*)d_ws;
    float* out = (float*)d_out;

    const bool merge = ws_size >= (size_t)(WS_IMG + NBATCH * NPIX) * sizeof(float);
    const int nzero = merge ? (WS_IMG + NBATCH * NPIX) : WS_IMG;
    pd_init_ws<<<(nzero + 255) / 256, 256, 0, stream>>>(ws, nzero);

    // 64 blocks * 8 waves = 512 waves, K-slice of 128 each (K = 65536)
    pd_mlp_gemm_wmma<<<64, 256, 0, stream>>>(x, w1, ws + WS_H);
    pd_mlp_finalize<<<1, 32, 0, stream>>>(ws + WS_H, b1, w2, b2, ws);

    const size_t ldsBytes = (size_t)NPIX * sizeof(float);  // 256 KB (CDNA5: 320 KB/WGP)
    if (merge) {
        (void)hipFuncSetAttribute(reinterpret_cast<const void*>(pd_scatter<8, true>),
                                  hipFuncAttributeMaxDynamicSharedMemorySize, (int)ldsBytes);
        pd_scatter<8, true><<<NBATCH * 8, 1024, ldsBytes, stream>>>(
            flow, values, rotations, shifts, inds, ws, out);
        pd_blur_affine<<<(NBATCH * NPIX + 255) / 256, 256, 0, stream>>>(ws, out);
    } else {
        (void)hipFuncSetAttribute(reinterpret_cast<const void*>(pd_scatter<1, false>),
                                  hipFuncAttributeMaxDynamicSharedMemorySize, (int)ldsBytes);
        pd_scatter<1, false><<<NBATCH, 1024, ldsBytes, stream>>>(
            flow, values, rotations, shifts, inds, ws, out);
    }
}